// MultiHeadAttentionQuantum_65481071409558
// MI455X (gfx1250) — compile-verified
//
#include <hip/hip_runtime.h>

// MI455X / gfx1250, wave32, WMMA f32 16x16x4.
//
// reference collapses analytically:
//   angles = x @ w_q^T + q_params          (GEMM1, K=768, N=8)
//   c_w    = cos(angle_w)
//   z_0    = c1*c2*...*c7 ; z_w = c0*...*c_w (w>=1)   (CNOT ring -> XOR parity)
//   out    = z @ w_out^T                   (GEMM2, K=8, N=768)

#define TOKENS 32768   // B*S = 16*2048
#define EMB    768
#define NW     8
#define WPB    8       // waves per block
#define ANG_STRIDE 20  // 16 + 4 pad -> conflict-free LDS transpose

typedef __attribute__((ext_vector_type(2))) float v2f;
typedef __attribute__((ext_vector_type(4))) float v4f;
typedef __attribute__((ext_vector_type(8))) float v8f;

__global__ __launch_bounds__(256)
void qmha_wmma_kernel(const float* __restrict__ x,
                      const float* __restrict__ wq,
                      const float* __restrict__ wout,
                      const float* __restrict__ qp,
                      float* __restrict__ out)
{
    __shared__ __align__(16) float angb[WPB][16 * ANG_STRIDE];
    __shared__ __align__(16) float zbuf[WPB][16 * NW];

    const unsigned tid  = threadIdx.x;
    const unsigned wv   = tid >> 5;
    const unsigned lane = tid & 31u;
    const unsigned lrow = lane & 15u;   // M row (A) or N col (B/C/D)
    const unsigned half = lane >> 4;    // 0 | 1
    const unsigned koff = half * 2u;    // K sub-offset inside a 4-step

    const unsigned tile = blockIdx.x * WPB + wv;
    const unsigned tok0 = tile * 16u;

    // ---------------- GEMM1: angles[16 tok][16 n] ----------------
    // A frag: lane -> x[tok0+lrow][k+koff .. k+koff+1]
    // B frag: lane -> w_q[n=lrow][k+koff .. k+koff+1]; lanes with n>=8 alias
    //         rows 0..7 -> garbage in N>=8 columns, never consumed.
    const float* xa = x  + (size_t)(tok0 + lrow) * EMB + koff;
    const float* wb = wq + (size_t)(lrow & 7u)  * EMB + koff;

    v8f acc0 = {0.f, 0.f, 0.f, 0.f, 0.f, 0.f, 0.f, 0.f};
    v8f acc1 = acc0;

    #pragma unroll 8
    for (int k = 0; k < EMB; k += 8) {
        v2f a0 = *(const v2f*)(xa + k);
        v2f b0 = *(const v2f*)(wb + k);
        acc0 = __builtin_amdgcn_wmma_f32_16x16x4_f32(
            false, a0, false, b0, (short)0, acc0, false, false);
        v2f a1 = *(const v2f*)(xa + k + 4);
        v2f b1 = *(const v2f*)(wb + k + 4);
        acc1 = __builtin_amdgcn_wmma_f32_16x16x4_f32(
            false, a1, false, b1, (short)0, acc1, false, false);
    }

    // angle += learned angle, take cos, transpose through LDS.
    // C/D layout: VGPR j -> row M = j + half*8, col N = lrow.
    const float qpl = qp[lrow & 7u];
    #pragma unroll
    for (int j = 0; j < 8; ++j) {
        float ang = acc0[j] + acc1[j] + qpl;
        angb[wv][(unsigned)(j + half * 8) * ANG_STRIDE + lrow] = __cosf(ang);
    }
    __syncthreads();

    // ---------------- classical collapse: prefix products ----------------
    if (lane < 16u) {
        const float* ar = &angb[wv][lrow * ANG_STRIDE];
        float c0 = ar[0], c1 = ar[1], c2 = ar[2], c3 = ar[3];
        float c4 = ar[4], c5 = ar[5], c6 = ar[6], c7 = ar[7];
        float p1 = c0 * c1;
        float p2 = p1 * c2;
        float p3 = p2 * c3;
        float p4 = p3 * c4;
        float p5 = p4 * c5;
        float p6 = p5 * c6;
        float p7 = p6 * c7;
        float t  = c1 * c2;
        t *= c3; t *= c4; t *= c5; t *= c6;
        float z0 = t * c7;          // ring-closing CNOT: wire 0 parity = b1^..^b7
        v4f lo = {z0, p1, p2, p3};
        v4f hi = {p4, p5, p6, p7};
        *(v4f*)&zbuf[wv][lrow * NW]     = lo;
        *(v4f*)&zbuf[wv][lrow * NW + 4] = hi;
    }
    __syncthreads();

    // ---------------- GEMM2: out[16 tok][768] = z @ w_out^T ----------------
    // A frags (K=8 -> two K=4 steps), resident for all 48 N-chunks.
    v2f za = *(const v2f*)&zbuf[wv][lrow * NW + koff];
    v2f zb = *(const v2f*)&zbuf[wv][lrow * NW + 4 + koff];

    float* ob = out + (size_t)(tok0 + half * 8u) * EMB + lrow;

    #pragma unroll 4
    for (int ec = 0; ec < EMB / 16; ++ec) {
        const int e0 = ec * 16;
        const float* wp = wout + (size_t)(e0 + (int)lrow) * NW;
        v2f b0 = *(const v2f*)(wp + koff);
        v2f b1 = *(const v2f*)(wp + 4 + koff);
        v8f d = {0.f, 0.f, 0.f, 0.f, 0.f, 0.f, 0.f, 0.f};
        d = __builtin_amdgcn_wmma_f32_16x16x4_f32(
            false, za, false, b0, (short)0, d, false, false);
        d = __builtin_amdgcn_wmma_f32_16x16x4_f32(
            false, zb, false, b1, (short)0, d, false, false);
        #pragma unroll
        for (int j = 0; j < 8; ++j) {
            ob[(size_t)j * EMB + e0] = d[j];
        }
    }
}

extern "C" void kernel_launch(void* const* d_in, const int* in_sizes, int n_in,
                              void* d_out, int out_size, void* d_ws, size_t ws_size,
                              hipStream_t stream) {
    (void)in_sizes; (void)n_in; (void)out_size; (void)d_ws; (void)ws_size;
    const float* x    = (const float*)d_in[0];   // [16,2048,768]
    const float* wq   = (const float*)d_in[1];   // [8,768]
    const float* wout = (const float*)d_in[2];   // [768,8]
    const float* qp   = (const float*)d_in[3];   // [8]
    float* out = (float*)d_out;                  // [16,2048,768]

    // 2048 tiles of 16 tokens; 8 waves (tiles) per 256-thread block.
    dim3 grid(TOKENS / (16 * WPB));   // 256
    dim3 block(256);
    qmha_wmma_kernel<<<grid, block, 0, stream>>>(x, wq, wout, qp, out);
}